// FalconH1Mixer_6133213298818
// MI455X (gfx1250) — compile-verified
//
#include <hip/hip_runtime.h>
#include <hip/hip_bf16.h>

typedef float v2f __attribute__((ext_vector_type(2)));
typedef float v8f __attribute__((ext_vector_type(8)));

#define HIDDEN   2048
#define INTER    4096
#define HEADS    64
#define HEAD_DIM 64
#define STATE    64
#define KCONV    4
#define CONV_DIM (INTER + 2 * STATE)          // 4224
#define PROJ     (INTER + CONV_DIM + HEADS)   // 8384
#define SEQ      2048

// ---------------------------------------------------------------------------
// CDNA5 async global->LDS copy (ASYNCcnt path). 16B per lane per issue.
// LDS destination: low 32 bits of a flat pointer to __shared__ = LDS offset.
// ---------------------------------------------------------------------------
__device__ __forceinline__ void async_copy_b128(void* lds_dst, const void* gsrc)
{
    unsigned lds = (unsigned)(size_t)lds_dst;
    asm volatile("global_load_async_to_lds_b128 %0, %1, off"
                 :: "v"(lds), "v"(gsrc)
                 : "memory");
}
// Wait until at most N async copies are outstanding (loads retire in order).
template <int N>
__device__ __forceinline__ void wait_async_le()
{
    asm volatile("s_wait_asynccnt %0" :: "i"(N) : "memory");
}

// ---------------------------------------------------------------------------
// Generic f32 GEMM-NT via V_WMMA_F32_16X16X4_F32:  C[M,N] = A[M,K] * B[N,K]^T
// Block = 128 threads (4 waves). Block tile 16(M) x 64(N), K-chunk 64,
// double-buffered async LDS staging (copies overlap WMMA compute).
// ---------------------------------------------------------------------------
#define KC 64
#define LSTR 68   // LDS row stride (floats): 16B-aligned rows, conflict-free b64

__global__ __launch_bounds__(128)
void gemm_nt_wmma(const float* __restrict__ A, const float* __restrict__ B,
                  float* __restrict__ C, int M, int N, int K,
                  int lda, int ldb, int ldc)
{
    __shared__ float As[2][16 * LSTR];
    __shared__ float Bs[2][64 * LSTR];

    const int t    = threadIdx.x;
    const int wv   = t >> 5;        // wave 0..3 -> N sub-tile
    const int lane = t & 31;
    const int hl   = lane >> 4;     // half-wave select
    const int lm   = lane & 15;     // row (A) / col (B,C)
    const int m0   = blockIdx.y * 16;
    const int n0   = blockIdx.x * 64;

    // 10 async b128 issues per thread per chunk (A:2 + B:8)
    auto stage = [&](int buf, int k0) {
        #pragma unroll
        for (int i = 0; i < 2; ++i) {
            int idx = t + i * 128;
            int row = idx >> 4, col = (idx & 15) * 4;
            async_copy_b128(&As[buf][row * LSTR + col],
                            A + (size_t)(m0 + row) * lda + k0 + col);
        }
        #pragma unroll
        for (int i = 0; i < 8; ++i) {
            int idx = t + i * 128;
            int row = idx >> 4, col = (idx & 15) * 4;
            async_copy_b128(&Bs[buf][row * LSTR + col],
                            B + (size_t)(n0 + row) * ldb + k0 + col);
        }
    };

    v8f acc = {};
    const int nk = K / KC;
    stage(0, 0);
    for (int kc = 0; kc < nk; ++kc) {
        const int buf = kc & 1;
        if (kc + 1 < nk) { stage(buf ^ 1, (kc + 1) * KC); wait_async_le<10>(); }
        else             { wait_async_le<0>(); }
        __syncthreads();

        // A frag (ISA 7.12.2, 32-bit 16x4): lane m = lm; VGPRs = K(kb),K(kb+1), kb=2*half
        // B frag (4x16, mirrored):          lane n = lm; VGPRs = K(kb),K(kb+1)
        const float* ap = &As[buf][lm * LSTR + hl * 2];
        const float* bp = &Bs[buf][(wv * 16 + lm) * LSTR + hl * 2];
        #pragma unroll
        for (int kk = 0; kk < KC; kk += 4) {
            v2f a = *(const v2f*)(ap + kk);
            v2f b = *(const v2f*)(bp + kk);
            acc = __builtin_amdgcn_wmma_f32_16x16x4_f32(
                false, a, false, b, (short)0, acc, false, false);
        }
        __syncthreads();
    }

    // C/D layout: VGPR i <-> row M = i + 8*half, col N = lm
    float* crow = C + (size_t)(m0 + hl * 8) * ldc + n0 + wv * 16 + lm;
    #pragma unroll
    for (int i = 0; i < 8; ++i)
        crow[(size_t)i * ldc] = acc[i];
}

// ---------------------------------------------------------------------------
// Causal depthwise conv (K=4) + SiLU over the hBC slice of proj.
// ---------------------------------------------------------------------------
__global__ void conv_silu_kernel(const float* __restrict__ proj,
                                 const float* __restrict__ conv_w,
                                 const float* __restrict__ conv_b,
                                 float* __restrict__ convs)
{
    int idx = blockIdx.x * blockDim.x + threadIdx.x;
    if (idx >= SEQ * CONV_DIM) return;
    int c = idx % CONV_DIM;
    int l = idx / CONV_DIM;
    float acc = conv_b[c];
    #pragma unroll
    for (int k = 0; k < KCONV; ++k) {
        int ll = l - (KCONV - 1) + k;
        if (ll >= 0)
            acc += proj[(size_t)ll * PROJ + INTER + c] * conv_w[c * KCONV + k];
    }
    convs[idx] = acc / (1.f + __expf(-acc));   // silu
}

// ---------------------------------------------------------------------------
// dt = softplus(proj_dt + dt_bias);  dA = exp(dt * (-exp(A_log)))
// ---------------------------------------------------------------------------
__global__ void dt_kernel(const float* __restrict__ proj,
                          const float* __restrict__ dt_bias,
                          const float* __restrict__ A_log,
                          float* __restrict__ dtp, float* __restrict__ dAb)
{
    int idx = blockIdx.x * blockDim.x + threadIdx.x;
    if (idx >= SEQ * HEADS) return;
    int h = idx & (HEADS - 1);
    int l = idx >> 6;
    float v  = proj[(size_t)l * PROJ + INTER + CONV_DIM + h] + dt_bias[h];
    float sp = (v > 20.f) ? v : log1pf(__expf(v));
    dtp[idx] = sp;
    dAb[idx] = __expf(sp * (-__expf(A_log[h])));
}

// ---------------------------------------------------------------------------
// Selective-state scan. Grid: 256 blocks = 64 heads x 4 state-slices (n of 16).
// Block: 128 threads; thread owns (p = t/2, 8 states). Double-buffered async
// LDS chunk staging overlapping the recurrent compute.
// ---------------------------------------------------------------------------
#define CLK 64
__global__ __launch_bounds__(128)
void scan_kernel(const float* __restrict__ convs, const float* __restrict__ dtp,
                 const float* __restrict__ dAb, float* __restrict__ ypart)
{
    __shared__ float Xc[2][CLK * 64];
    __shared__ float Bc[2][CLK * 16];
    __shared__ float Cc[2][CLK * 16];
    __shared__ float dAc[2][CLK];
    __shared__ float dtc[2][CLK];

    const int h  = blockIdx.x & 63;
    const int nb = blockIdx.x >> 6;
    const int n0 = nb * 16;
    const int t  = threadIdx.x;
    const int p  = t >> 1;
    const int j  = t & 1;

    // 12 async b128 issues per thread per chunk (X:8 + B:2 + C:2)
    auto stage = [&](int buf, int l0) {
        #pragma unroll
        for (int i = 0; i < 8; ++i) {
            int idx = t + i * 128;
            int ll = idx >> 4, pp = (idx & 15) * 4;
            async_copy_b128(&Xc[buf][ll * 64 + pp],
                            convs + (size_t)(l0 + ll) * CONV_DIM + h * 64 + pp);
        }
        #pragma unroll
        for (int i = 0; i < 2; ++i) {
            int idx = t + i * 128;
            int ll = idx >> 2, nn = (idx & 3) * 4;
            async_copy_b128(&Bc[buf][ll * 16 + nn],
                            convs + (size_t)(l0 + ll) * CONV_DIM + INTER + n0 + nn);
            async_copy_b128(&Cc[buf][ll * 16 + nn],
                            convs + (size_t)(l0 + ll) * CONV_DIM + INTER + STATE + n0 + nn);
        }
        if (t < CLK)          dAc[buf][t]       = dAb[(size_t)(l0 + t) * HEADS + h];
        else if (t < 2 * CLK) dtc[buf][t - CLK] = dtp[(size_t)(l0 + t - CLK) * HEADS + h];
    };

    float S[8];
    #pragma unroll
    for (int i = 0; i < 8; ++i) S[i] = 0.f;

    float* yout = ypart + (size_t)nb * SEQ * INTER + (size_t)h * 64 + p;

    const int nch = SEQ / CLK;   // 32 chunks
    stage(0, 0);
    for (int c = 0; c < nch; ++c) {
        const int buf = c & 1;
        const int l0  = c * CLK;
        if (c + 1 < nch) { stage(buf ^ 1, l0 + CLK); wait_async_le<12>(); }
        else             { wait_async_le<0>(); }
        __syncthreads();

        for (int ll = 0; ll < CLK; ++ll) {
            float da = dAc[buf][ll];
            float a  = dtc[buf][ll] * Xc[buf][ll * 64 + p];
            // guaranteed ds_load_b128 fragments of B/C
            const float4* Bp = (const float4*)&Bc[buf][ll * 16 + j * 8];
            const float4* Cp = (const float4*)&Cc[buf][ll * 16 + j * 8];
            float4 b0 = Bp[0], b1 = Bp[1];
            float4 c0 = Cp[0], c1 = Cp[1];
            float accy;
            S[0] = da * S[0] + a * b0.x;  accy  = S[0] * c0.x;
            S[1] = da * S[1] + a * b0.y;  accy += S[1] * c0.y;
            S[2] = da * S[2] + a * b0.z;  accy += S[2] * c0.z;
            S[3] = da * S[3] + a * b0.w;  accy += S[3] * c0.w;
            S[4] = da * S[4] + a * b1.x;  accy += S[4] * c1.x;
            S[5] = da * S[5] + a * b1.y;  accy += S[5] * c1.y;
            S[6] = da * S[6] + a * b1.z;  accy += S[6] * c1.z;
            S[7] = da * S[7] + a * b1.w;  accy += S[7] * c1.w;
            accy += __shfl_xor(accy, 1);      // wave32: pair lanes share p
            if (j == 0) yout[(size_t)(l0 + ll) * INTER] = accy;
        }
        __syncthreads();
    }
}

// ---------------------------------------------------------------------------
// Combine 4 y-partials + D*x, grouped RMSNorm over 4096, * silu(gate).
// Result overwrites the gate region of proj (in-place, same-thread RAW).
// Thread handles 16 contiguous channels -> float4 global traffic.
// ---------------------------------------------------------------------------
__global__ __launch_bounds__(256)
void combine_norm_gate(const float* __restrict__ ypart,
                       const float* __restrict__ convs,
                       const float* __restrict__ Dv,
                       const float* __restrict__ norm_w,
                       float* __restrict__ proj)
{
    __shared__ float red[8];
    const int l  = blockIdx.x;
    const int t  = threadIdx.x;
    const int c0 = t * 16;
    const float d = Dv[c0 >> 6];   // 16 contiguous channels sit in one head

    float4 yv[4];
    #pragma unroll
    for (int i = 0; i < 4; ++i) {
        float4 x = *(const float4*)(convs + (size_t)l * CONV_DIM + c0 + i * 4);
        yv[i].x = d * x.x; yv[i].y = d * x.y; yv[i].z = d * x.z; yv[i].w = d * x.w;
    }
    #pragma unroll
    for (int q = 0; q < 4; ++q) {
        const float* yp = ypart + ((size_t)q * SEQ + l) * INTER + c0;
        #pragma unroll
        for (int i = 0; i < 4; ++i) {
            float4 v = *(const float4*)(yp + i * 4);
            yv[i].x += v.x; yv[i].y += v.y; yv[i].z += v.z; yv[i].w += v.w;
        }
    }
    float ss = 0.f;
    #pragma unroll
    for (int i = 0; i < 4; ++i)
        ss += yv[i].x * yv[i].x + yv[i].y * yv[i].y
            + yv[i].z * yv[i].z + yv[i].w * yv[i].w;

    #pragma unroll
    for (int o = 16; o > 0; o >>= 1) ss += __shfl_xor(ss, o);
    if ((t & 31) == 0) red[t >> 5] = ss;
    __syncthreads();
    if (t < 8) {
        float v = red[t];
        v += __shfl_xor(v, 4);
        v += __shfl_xor(v, 2);
        v += __shfl_xor(v, 1);
        if (t == 0) red[0] = v;
    }
    __syncthreads();
    const float rs = rsqrtf(red[0] / (float)INTER + 1e-6f);

    #pragma unroll
    for (int i = 0; i < 4; ++i) {
        float4 g  = *(const float4*)(proj + (size_t)l * PROJ + c0 + i * 4);
        float4 nw = *(const float4*)(norm_w + c0 + i * 4);
        float4 o;
        o.x = yv[i].x * rs * nw.x * (g.x / (1.f + __expf(-g.x)));
        o.y = yv[i].y * rs * nw.y * (g.y / (1.f + __expf(-g.y)));
        o.z = yv[i].z * rs * nw.z * (g.z / (1.f + __expf(-g.z)));
        o.w = yv[i].w * rs * nw.w * (g.w / (1.f + __expf(-g.w)));
        *(float4*)(proj + (size_t)l * PROJ + c0 + i * 4) = o;
    }
}

// ---------------------------------------------------------------------------
extern "C" void kernel_launch(void* const* d_in, const int* in_sizes, int n_in,
                              void* d_out, int out_size, void* d_ws, size_t ws_size,
                              hipStream_t stream)
{
    const float* hidden     = (const float*)d_in[0];  // [2048, 2048]
    const float* in_proj_w  = (const float*)d_in[1];  // [8384, 2048]
    const float* conv_w     = (const float*)d_in[2];  // [4224, 4]
    const float* conv_b     = (const float*)d_in[3];  // [4224]
    const float* dt_bias    = (const float*)d_in[4];  // [64]
    const float* A_log      = (const float*)d_in[5];  // [64]
    const float* Dv         = (const float*)d_in[6];  // [64]
    const float* norm_w     = (const float*)d_in[7];  // [4096]
    const float* out_proj_w = (const float*)d_in[8];  // [2048, 4096]
    float*       out        = (float*)d_out;          // [2048, 2048]

    float* ws = (float*)d_ws;
    float* proj  = ws;                                         // 2048*8384
    float* convs = proj  + (size_t)SEQ * PROJ;                 // 2048*4224
    float* dtp   = convs + (size_t)SEQ * CONV_DIM;             // 2048*64
    float* dAb   = dtp   + (size_t)SEQ * HEADS;                // 2048*64
    float* ypart = dAb   + (size_t)SEQ * HEADS;                // 4*2048*4096

    // 1) in_proj: proj = hidden @ in_proj_w^T   (M=2048, N=8384, K=2048)
    gemm_nt_wmma<<<dim3(PROJ / 64, SEQ / 16), 128, 0, stream>>>(
        hidden, in_proj_w, proj, SEQ, PROJ, HIDDEN, HIDDEN, HIDDEN, PROJ);

    // 2) causal conv + silu over hBC
    {
        int n = SEQ * CONV_DIM;
        conv_silu_kernel<<<(n + 255) / 256, 256, 0, stream>>>(proj, conv_w, conv_b, convs);
    }

    // 3) dt softplus / dA
    {
        int n = SEQ * HEADS;
        dt_kernel<<<(n + 255) / 256, 256, 0, stream>>>(proj, dt_bias, A_log, dtp, dAb);
    }

    // 4) selective scan (64 heads x 4 state slices)
    scan_kernel<<<256, 128, 0, stream>>>(convs, dtp, dAb, ypart);

    // 5) combine + RMSNorm + gate  (writes into gate region of proj)
    combine_norm_gate<<<SEQ, 256, 0, stream>>>(ypart, convs, Dv, norm_w, proj);

    // 6) out_proj: out = ynorm @ out_proj_w^T  (M=2048, N=2048, K=4096, lda=8384)
    gemm_nt_wmma<<<dim3(HIDDEN / 64, SEQ / 16), 128, 0, stream>>>(
        proj, out_proj_w, out, SEQ, HIDDEN, INTER, PROJ, INTER, HIDDEN);
}